// SimonConstantVelocityModel_83021717832571
// MI455X (gfx1250) — compile-verified
//
#include <hip/hip_runtime.h>
#include <hip/hip_bf16.h>
#include <stdint.h>

// ---------------------------------------------------------------------------
// SimonConstantVelocityModel log-likelihood on gfx1250 (MI455X, wave32)
//
//   LL = E*beta - sum_e ||dz + dv*t_e||^2  -  sum_{i<j} integ(i,j)
//
// Kernel 1: event sum (random gathers served from LDS-staged point tables)
// Kernel 2: triangular all-pairs closed-form integral (transcendental bound),
//           balanced by pairing row r with row P-1-r (exactly P-1 pairs each)
// Kernel 3: deterministic finalize
//
// CDNA5-specific paths:
//   * V_WMMA_F32_16X16X4_F32 block reduction (A = ones => column sums of B)
//   * GLOBAL_LOAD_ASYNC_TO_LDS_B128 staging (ASYNCcnt + s_wait_asynccnt)
//   * wave32 ds_bpermute shuffle tail
// ---------------------------------------------------------------------------

#define N_POINTS 4096

typedef float v2f __attribute__((ext_vector_type(2)));
typedef float v8f __attribute__((ext_vector_type(8)));

// Block-wide sum of 256 floats. `red` must be 256-float shared scratch.
// Result valid on threadIdx.x == 0 only.
__device__ __forceinline__ float block_reduce_wmma(float val, float* red) {
    const int tid = threadIdx.x;
    __syncthreads();              // red may alias the LDS point table
    red[tid] = val;
    __syncthreads();
    float out = 0.0f;
    if (tid < 32) {               // wave 0: EXEC all-1s (WMMA requirement)
        const int lane = tid;
        v2f ones; ones.x = 1.0f; ones.y = 1.0f;   // every A[m][k] = 1
        v8f acc = {};
#pragma unroll
        for (int c = 0; c < 4; ++c) {
            v2f bv;
            bv.x = red[c * 64 + lane];
            bv.y = red[c * 64 + 32 + lane];
            // D = A(ones) x B + C -> column sums of B accumulated into C.
            acc = __builtin_amdgcn_wmma_f32_16x16x4_f32(
                false, ones, false, bv, (short)0, acc, false, false);
        }
        // Row 0 of D: lanes 0..15, VGPR0 hold the 16 column sums.
        float v = acc[0];
        v += __shfl_down(v, 8);
        v += __shfl_down(v, 4);
        v += __shfl_down(v, 2);
        v += __shfl_down(v, 1);
        out = v;                  // lane 0 == total of all 256 inputs
    }
    return out;
}

// Async DMA of z0/v0 (32 KB each) straight into LDS, bypassing VGPRs.
// GVS mode: saddr = uniform table base, vaddr = per-lane 32-bit byte offset.
__device__ __forceinline__ void stage_tables_async(const float2* __restrict__ z0,
                                                   const float2* __restrict__ v0,
                                                   float2* zs, float2* vs) {
    const uint32_t lz = (uint32_t)(uintptr_t)zs;   // low 32 bits = LDS offset
    const uint32_t lv = (uint32_t)(uintptr_t)vs;
    for (int k = threadIdx.x; k < N_POINTS / 2; k += blockDim.x) {
        const uint32_t off = (uint32_t)k * 16u;    // 16 B (two float2) per lane
        const uint32_t dz = lz + off;
        const uint32_t dv = lv + off;
        asm volatile("global_load_async_to_lds_b128 %0, %1, %2"
                     :: "v"(dz), "v"(off), "s"(z0) : "memory");
        asm volatile("global_load_async_to_lds_b128 %0, %1, %2"
                     :: "v"(dv), "v"(off), "s"(v0) : "memory");
    }
    asm volatile("s_wait_asynccnt 0" ::: "memory");
    __syncthreads();
}

// ---- Kernel 1: sum_e ||(z_u - z_v) + (v_u - v_v) * t_e||^2 ------------------
__global__ void scvm_events_kernel(const int* __restrict__ eu,
                                   const int* __restrict__ ev,
                                   const float* __restrict__ et,
                                   const float2* __restrict__ z0,
                                   const float2* __restrict__ v0,
                                   float* __restrict__ ws_event, int E) {
    __shared__ float2 zs[N_POINTS];           // 32 KB
    __shared__ float2 vs[N_POINTS];           // 32 KB (of 320 KB/WGP)
    float* red = (float*)zs;                  // reused after the main loop
    stage_tables_async(z0, v0, zs, vs);

    float s = 0.0f;
    const int stride = gridDim.x * blockDim.x;
    for (int e = blockIdx.x * blockDim.x + threadIdx.x; e < E; e += stride) {
        const int u = eu[e];
        const int w = ev[e];
        const float t = et[e];
        const float2 zu = zs[u], zv = zs[w];
        const float2 vu = vs[u], vv = vs[w];
        const float dx = (zu.x - zv.x) + (vu.x - vv.x) * t;
        const float dy = (zu.y - zv.y) + (vu.y - vv.y) * t;
        s += dx * dx + dy * dy;
    }
    const float r = block_reduce_wmma(s, red);
    if (threadIdx.x == 0) ws_event[blockIdx.x] = r;
}

// ---- Kernel 2: sum over unordered pairs i<j of integ(i,j) -------------------
// Balanced triangular mapping: superrow r covers rows {r, P-1-r}; both halves
// together hold exactly P-1 column slots, so TOT = (P/2)*(P-1) = C(P,2).
__global__ void scvm_pairs_kernel(const float* __restrict__ t0p,
                                  const float* __restrict__ tnp,
                                  const float* __restrict__ betap,
                                  const float2* __restrict__ z0,
                                  const float2* __restrict__ v0,
                                  float* __restrict__ ws_pair) {
    __shared__ float2 zs[N_POINTS];
    __shared__ float2 vs[N_POINTS];
    float* red = (float*)zs;
    stage_tables_async(z0, v0, zs, vs);

    const float b  = betap[0];
    const float T0 = t0p[0];
    const float T1 = tnp[0];
    const float sqrt_pi = 1.7724538509055159f;

    float s = 0.0f;
    const int COLS = N_POINTS - 1;            // 4095
    const int TOT  = (N_POINTS / 2) * COLS;   // 8,386,560 = C(4096,2)
    const int stride = gridDim.x * blockDim.x;
    for (int idx = blockIdx.x * blockDim.x + threadIdx.x; idx < TOT; idx += stride) {
        const int r = idx / COLS;             // strength-reduced to mulhi
        const int c = idx - r * COLS;
        const int row1len = N_POINTS - 1 - r; // pairs in row r
        int i, j;
        if (c < row1len) { i = r;                j = r + 1 + c; }
        else             { i = N_POINTS - 1 - r; j = N_POINTS - r + (c - row1len); }

        const float2 zi = zs[i], zj = zs[j];
        const float2 vi = vs[i], vj = vs[j];
        const float a  = zi.x - zj.x;
        const float bb = zi.y - zj.y;
        const float m  = vi.x - vj.x;
        const float n  = vi.y - vj.y;
        const float s2 = m * m + n * n;
        const float inv_s2 = 1.0f / s2;
        const float expo = ((b - bb * bb) * m * m + 2.0f * a * bb * m * n
                            - n * n * (a * a - b)) * inv_s2;
        const float am = a * m + bb * n;
        const float inv_s = 1.0f / sqrtf(s2);
        const float e0 = erff((s2 * T0 + am) * inv_s);
        const float e1 = erff((s2 * T1 + am) * inv_s);
        s += -sqrt_pi * expf(expo) * (e0 - e1) * (0.5f * inv_s);
    }
    const float r2 = block_reduce_wmma(s, red);
    if (threadIdx.x == 0) ws_pair[blockIdx.x] = r2;
}

// ---- Kernel 3: deterministic finalize --------------------------------------
__global__ void scvm_finalize_kernel(const float* __restrict__ ws,
                                     const float* __restrict__ betap,
                                     float* __restrict__ out,
                                     int nbe, int nbp, float Ef) {
    __shared__ float red[256];
    float c = 0.0f;
    for (int k = threadIdx.x; k < nbe; k += 256) c -= ws[k];        // -sum d2
    for (int k = threadIdx.x; k < nbp; k += 256) c -= ws[nbe + k];  // -non_event
    const float r = block_reduce_wmma(c, red);
    if (threadIdx.x == 0) out[0] = Ef * betap[0] + r;
}

extern "C" void kernel_launch(void* const* d_in, const int* in_sizes, int n_in,
                              void* d_out, int out_size, void* d_ws, size_t ws_size,
                              hipStream_t stream) {
    const int*    eu   = (const int*)d_in[0];
    const int*    ev   = (const int*)d_in[1];
    const float*  et   = (const float*)d_in[2];
    const float*  t0   = (const float*)d_in[3];
    const float*  tn   = (const float*)d_in[4];
    const float*  beta = (const float*)d_in[5];
    const float2* z0   = (const float2*)d_in[6];
    const float2* v0   = (const float2*)d_in[7];
    float*        out  = (float*)d_out;
    float*        ws   = (float*)d_ws;

    const int E   = in_sizes[0];
    const int NBE = 512;    // event partials
    const int NBP = 2048;   // pair partials

    scvm_events_kernel<<<NBE, 256, 0, stream>>>(eu, ev, et, z0, v0, ws, E);
    scvm_pairs_kernel<<<NBP, 256, 0, stream>>>(t0, tn, beta, z0, v0, ws + NBE);
    scvm_finalize_kernel<<<1, 256, 0, stream>>>(ws, beta, out, NBE, NBP, (float)E);
}